// ContinuousAxialDW_26302379720990
// MI455X (gfx1250) — compile-verified
//
#include <hip/hip_runtime.h>
#include <stdint.h>

// Problem constants (from the reference)
#define BN 8
#define CN 96
#define HN 256
#define WN 256
#define KT 7

// Tiling
#define PAD     8                    // halo (rows and cols); valid for r <= 8/3 (harness: r=1.5)
#define TILE_H  32                   // output rows per workgroup
#define LROWS   (TILE_H + 2 * PAD)   // 48 staged rows
#define WP      (WN + 2 * PAD)       // 272 floats/row: 8-float zero pad each side
#define NTHR    256                  // 8 wave32
#define NOFF    17                   // combined integer tap offsets o in [-8, 8]

typedef float    f32x4 __attribute__((ext_vector_type(4)));
typedef uint32_t u32x4 __attribute__((ext_vector_type(4)));
typedef int      i32x4 __attribute__((ext_vector_type(4)));
typedef int      i32x8 __attribute__((ext_vector_type(8)));

__global__ __launch_bounds__(NTHR)
void axial_dw_kernel(const float* __restrict__ x,
                     const float* __restrict__ wh,   // [C,1,KT,1] -> [c*KT + i]
                     const float* __restrict__ ww,   // [C,1,1,KT] -> [c*KT + i]
                     const float* __restrict__ rp,   // scalar r
                     float* __restrict__ out)
{
    __shared__ float tile[LROWS * WP];               // 51 KB

    const int t       = threadIdx.x;
    const int bid     = blockIdx.x;
    const int tile_id = bid & 7;                     // 8 row-tiles per plane
    const int plane   = bid >> 3;                    // b*CN + c
    const int c       = plane % CN;
    const int h0      = tile_id * TILE_H;

    const float* __restrict__ xplane = x + (size_t)plane * (HN * WN);

    // ---- zero the left/right row pads (48 rows * 16 floats = 768) ----
    #pragma unroll
    for (int k2 = 0; k2 < 3; ++k2) {
        const int idx = t + k2 * NTHR;               // 0..767
        const int row = idx >> 4;
        const int j   = idx & 15;
        const int col = (j < PAD) ? j : (WN + j);    // 0..7 or 264..271
        tile[row * WP + col] = 0.0f;
    }
    // ---- zero the top out-of-image halo rows (first tile only) ----
    if (h0 == 0) {
        #pragma unroll
        for (int k2 = 0; k2 < 8; ++k2) {
            const int idx = t + k2 * NTHR;           // rows 0..7, cols 0..255
            tile[(idx >> 8) * WP + PAD + (idx & 255)] = 0.0f;
        }
    }

    const int start_row  = (h0 - PAD < 0) ? 0 : (h0 - PAD);
    const int lds_row0   = start_row - (h0 - PAD);   // 8 for first tile, else 0
    const int load_rows  = LROWS - lds_row0;         // rows the DMA writes
    const int rows_avail = HN - start_row;           // TDM zero-fills rows beyond this

#if __has_builtin(__builtin_amdgcn_tensor_load_to_lds)
    // ---------- Tensor Data Mover: one 2D-tile descriptor per workgroup ----------
    if (t == 0) {
        const uint64_t ga = (uint64_t)(uintptr_t)(xplane + (size_t)start_row * WN);
        const uint32_t lds_addr = (uint32_t)(uintptr_t)(&tile[lds_row0 * WP + PAD]);
        u32x4 g0;
        g0[0] = 1u;                                   // count=1, no gather, user mode
        g0[1] = lds_addr;                             // D#.lds_addr (bytes)
        g0[2] = (uint32_t)ga;                         // global_addr[31:0]
        g0[3] = ((uint32_t)(ga >> 32) & 0x01FFFFFFu)  // global_addr[56:32]
              | (2u << 30);                           // type = 2 ("image")
        i32x8 g1;
        g1[0] = (int)((2u << 16)                      // data_size = 4B
                    | (1u << 20)                      // pad_enable (build row pads in LDS)
                    | (7u << 22)                      // pad_interval: every 256 dwords
                    | (15u << 25));                   // pad_amount: 16 dwords
        g1[1] = (int)((uint32_t)WN << 16);            // tensor_dim0[15:0] = 256
        g1[2] = (int)(((uint32_t)rows_avail) << 16);  // tensor_dim1[15:0] (dim0 hi = 0)
        g1[3] = (int)((uint32_t)WN << 16);            // tile_dim0 = 256 (dim1 hi = 0)
        g1[4] = load_rows;                            // tile_dim1 (tile_dim2=0 -> 2D)
        g1[5] = WN;                                   // tensor_dim0_stride lo = 256
        g1[6] = 0;                                    // stride hi / dim1_stride (unused, 2D)
        g1[7] = 0;
        const i32x4 z4 = {0, 0, 0, 0};
        const i32x8 z8 = {0, 0, 0, 0, 0, 0, 0, 0};
        __builtin_amdgcn_tensor_load_to_lds(g0, g1, z4, z4, z8, 0);
    }
#else
    // ---------- fallback: per-lane async copies ----------
    #pragma unroll
    for (int k2 = 0; k2 < 12; ++k2) {
        const int chunk = t + k2 * NTHR;             // 0..3071
        const int lrow  = chunk >> 6;
        const int col4  = (chunk & 63) << 2;
        const int gh    = h0 - PAD + lrow;
        const uint32_t lds_addr =
            (uint32_t)(uintptr_t)(&tile[lrow * WP + PAD + col4]);
        if (gh >= 0 && gh < HN) {
            const uint64_t gaddr =
                (uint64_t)(uintptr_t)(xplane + (size_t)gh * WN + col4);
            asm volatile("global_load_async_to_lds_b128 %0, %1, off"
                         :: "v"(lds_addr), "v"(gaddr) : "memory");
        } else {
            f32x4 zz = {0.f, 0.f, 0.f, 0.f};
            *(f32x4*)(&tile[lrow * WP + PAD + col4]) = zz;
        }
    }
#endif

    // ---------- wave-uniform combined-tap setup (overlaps the DMA) ----------
    // floor(p + off*r) = p + floor(off*r): each axis collapses to integer-offset taps
    // cw[o] = sum_i [dd_i==o] w_i*(1-f_i) + [dd_i+1==o] w_i*f_i,  o in [-8,8].
    // Boundary masks are replaced by the zero padding staged into LDS.
    float r = rp[0];
    r = (r < 1.0f) ? 1.0f : r;                       // clamp(r, min=1)
    float thv[KT], twv[KT], frv[KT];
    int   dv[KT];
    #pragma unroll
    for (int i = 0; i < KT; ++i) {
        thv[i] = wh[c * KT + i];
        twv[i] = ww[c * KT + i];
        const float d  = (float)(i - KT / 2) * r;
        const float fl = floorf(d);
        int o = (int)fl;
        o = (o < -PAD) ? -PAD : ((o > PAD - 1) ? (PAD - 1) : o);
        dv[i]  = o;
        frv[i] = d - fl;
    }
    float chv[NOFF], cwv[NOFF];                      // constant-indexed -> stays in regs
    #pragma unroll
    for (int o = 0; o < NOFF; ++o) {
        float a = 0.f, b = 0.f;
        #pragma unroll
        for (int i = 0; i < KT; ++i) {
            const float w0 = 1.0f - frv[i];
            a += (dv[i] + 8 == o) ? thv[i] * w0 : 0.f;
            a += (dv[i] + 9 == o) ? thv[i] * frv[i] : 0.f;
            b += (dv[i] + 8 == o) ? twv[i] * w0 : 0.f;
            b += (dv[i] + 9 == o) ? twv[i] * frv[i] : 0.f;
        }
        chv[o] = a;
        cwv[o] = b;
    }

#if __has_builtin(__builtin_amdgcn_tensor_load_to_lds)
    __builtin_amdgcn_s_wait_tensorcnt(0);            // drain TDM (issuing wave; others no-op)
#else
    asm volatile("s_wait_asynccnt 0x0" ::: "memory");
#endif
    __syncthreads();                                 // also covers DScnt zero-fills

    // ---------- compute: 4-wide x 8-row microtile per thread ----------
    float* __restrict__ oplane = out + (size_t)plane * (HN * WN);
    const int w4   = (t & 63) << 2;                  // column group
    const int rowg = t >> 6;                         // 4 row groups of 8 rows

    for (int j = 0; j < 8; ++j) {
        const int k  = rowg * 8 + j;
        const int lr = k + PAD;
        const float* rowp = &tile[lr * WP + PAD + w4];

        // Horizontal window [w4-8, w4+12): 5 aligned b128 LDS loads -> registers.
        // All horizontal taps become constant-index register extracts (o is unrolled).
        f32x4 wv[5];
        #pragma unroll
        for (int q = 0; q < 5; ++q)
            wv[q] = *(const f32x4*)(rowp - 8 + 4 * q);
        float win[20];
        #pragma unroll
        for (int q = 0; q < 5; ++q) {
            win[4 * q + 0] = wv[q][0];
            win[4 * q + 1] = wv[q][1];
            win[4 * q + 2] = wv[q][2];
            win[4 * q + 3] = wv[q][3];
        }

        // identity term + vertical o==0 tap reuse the center registers
        f32x4 acc = wv[2] * (1.0f + chv[PAD]);

        // vertical taps o != 0: one b128 per tap serves 4 pixels; uniform zero-skip
        #pragma unroll
        for (int o = -PAD; o <= PAD; ++o) {
            if (o == 0) continue;
            const float wgt = chv[o + PAD];
            if (wgt != 0.0f) {
                const f32x4 v = *(const f32x4*)(rowp + o * WP);
                acc += v * wgt;
            }
        }
        // horizontal taps: pure register fmas (zero-weight taps cost one fma, no branch)
        #pragma unroll
        for (int o = -PAD; o <= PAD; ++o) {
            const float wgt = cwv[o + PAD];
            acc[0] = fmaf(win[o + PAD + 0], wgt, acc[0]);
            acc[1] = fmaf(win[o + PAD + 1], wgt, acc[1]);
            acc[2] = fmaf(win[o + PAD + 2], wgt, acc[2]);
            acc[3] = fmaf(win[o + PAD + 3], wgt, acc[3]);
        }

        const int h = h0 + k;
        __builtin_nontemporal_store(acc, (f32x4*)&oplane[(size_t)h * WN + w4]);
    }
}

extern "C" void kernel_launch(void* const* d_in, const int* in_sizes, int n_in,
                              void* d_out, int out_size, void* d_ws, size_t ws_size,
                              hipStream_t stream)
{
    const float* x  = (const float*)d_in[0];   // [B,C,H,W]
    const float* wh = (const float*)d_in[1];   // [C,1,KH,1]
    const float* ww = (const float*)d_in[2];   // [C,1,1,KW]
    const float* rp = (const float*)d_in[3];   // scalar r
    float* outp     = (float*)d_out;

    const int nblocks = BN * CN * (HN / TILE_H);   // 8*96*8 = 6144 workgroups
    axial_dw_kernel<<<dim3(nblocks), dim3(NTHR), 0, stream>>>(x, wh, ww, rp, outp);
}